// SparseGraphConv_82102594830695
// MI455X (gfx1250) — compile-verified
//
#include <hip/hip_runtime.h>

#define N_NODES 100000
#define N_EDGES 1600000
#define IN_F 256
#define OUT_F 128

typedef __attribute__((ext_vector_type(2))) float v2f;
typedef __attribute__((ext_vector_type(8))) float v8f;
typedef __attribute__((ext_vector_type(4))) unsigned int u32x4;
typedef __attribute__((ext_vector_type(8))) int i32x8;
typedef __attribute__((ext_vector_type(4))) int i32x4;

__device__ __forceinline__ void atomic_add_f32(float* p, float v) {
    __hip_atomic_fetch_add(p, v, __ATOMIC_RELAXED, __HIP_MEMORY_SCOPE_AGENT);
}

// Raw LDS byte offset of a __shared__ pointer (flat LDS aperture: addr[31:0] = offset).
__device__ __forceinline__ unsigned lds_off(const void* p) {
    return (unsigned)(unsigned long long)(uintptr_t)p;
}

// ---------------------------------------------------------------- TDM 2D tile load
// Builds D# group0/group1 per CDNA5 ISA ch.8 and issues TENSOR_LOAD_TO_LDS.
//  - data_size = 4 bytes (fp32)
//  - pad_enable: after every (2<<pad_code) DWORDs insert (pad_amt+1) DWORDs pad
//  - OOB rows (y >= tensor_d1) read as zero (hardware zero-fill)
__device__ __forceinline__ void tdm_load_2d(unsigned lds_addr, const void* gaddr,
                                            unsigned tensor_d0, unsigned tensor_d1,
                                            unsigned tile_d0, unsigned tile_d1,
                                            unsigned d0_stride,
                                            unsigned pad_code, unsigned pad_amt) {
    unsigned long long ga = (unsigned long long)(uintptr_t)gaddr;
    u32x4 g0;
    g0[0] = 1u;                                             // count=1, user descriptor
    g0[1] = lds_addr;                                       // LDS byte address
    g0[2] = (unsigned)(ga & 0xFFFFFFFFu);                   // global_addr[31:0]
    g0[3] = (unsigned)((ga >> 32) & 0x01FFFFFFu) | (2u << 30); // global_addr[56:32] | type=2
    i32x8 g1;
    g1[0] = (int)((2u << 16)                                // data_size = 4B
                | (1u << 20)                                // pad_enable
                | (pad_code << 22)                          // pad_interval
                | (pad_amt << 25));                         // pad_amount
    g1[1] = (int)(tensor_d0 << 16);                         // tensor_dim0[15:0]
    g1[2] = (int)((tensor_d0 >> 16) | (tensor_d1 << 16));   // dim0 hi | tensor_dim1 lo
    g1[3] = (int)((tensor_d1 >> 16) | (tile_d0 << 16));     // dim1 hi | tile_dim0
    g1[4] = (int)(tile_d1 & 0xFFFFu);                       // tile_dim1, tile_dim2=0
    g1[5] = (int)d0_stride;                                 // tensor_dim0_stride[31:0]
    g1[6] = 0;
    g1[7] = 0;
    i32x4 z4 = {0, 0, 0, 0};
#if __has_include(<hip/amd_detail/amd_gfx1250_TDM.h>)
    i32x8 z8 = {0, 0, 0, 0, 0, 0, 0, 0};
    __builtin_amdgcn_tensor_load_to_lds(g0, g1, z4, z4, z8, 0);
#else
    __builtin_amdgcn_tensor_load_to_lds(g0, g1, z4, z4, 0);
#endif
}

// ---------------------------------------------------------------- zero fill
__global__ void zero_kernel(float* __restrict__ a, long long n) {
    long long i = (long long)blockIdx.x * blockDim.x + threadIdx.x;
    long long stride = (long long)gridDim.x * blockDim.x;
    for (; i < n; i += stride) a[i] = 0.0f;
}

// ---------------------------------------------------------------- degrees
__global__ void degree_kernel(const int* __restrict__ src, const int* __restrict__ dst,
                              float* __restrict__ deg_out, float* __restrict__ deg_in) {
    int e = blockIdx.x * blockDim.x + threadIdx.x;
    if (e < N_EDGES) {
        atomic_add_f32(&deg_out[src[e]], 1.0f);
        atomic_add_f32(&deg_in[dst[e]], 1.0f);
    }
}

// ---------------------------------------------------------------- WMMA GEMM
// h[m, :] = (feat[m, :] * norm_src[m]) @ W     (M=100000, K=256, N=128)
// block: 256 threads = 8 waves; block tile = 128 rows x 128 cols.
// Panels staged by the Tensor Data Mover; TDM pad fields create the
// bank-conflict-free LDS strides in hardware.
#define KP 32        // K panel per iteration
#define FS 36        // feat LDS row stride: 32 + TDM pad 4  -> (36m+k)%64 conflict-free
#define WSTR 132     // weight LDS row stride: 128 + TDM pad 4

__global__ __launch_bounds__(256) void gemm_kernel(const float* __restrict__ feat,
                                                   const float* __restrict__ weight,
                                                   const float* __restrict__ deg_out,
                                                   float* __restrict__ h) {
    __shared__ float lds_feat[128 * FS];   // 18.0 KB
    __shared__ float lds_w[KP * WSTR];     // 16.5 KB

    const int tid  = threadIdx.x;
    const int wave = tid >> 5;
    const int lane = tid & 31;
    const int l15  = lane & 15;
    const int koff = (lane >> 4) << 1;     // lanes 0-15: K+0,1 ; lanes 16-31: K+2,3
    const int m0   = blockIdx.x * 128;

    // per-lane A row and its src-degree norm (folded into A fragments)
    const int arow = (wave << 4) + l15;
    const int gm   = m0 + arow;
    float nrm = 0.0f;
    if (gm < N_NODES) {
        float d = deg_out[gm];
        nrm = rsqrtf(d < 1.0f ? 1.0f : d);
    }
    const unsigned rows_left = (unsigned)(N_NODES - m0 < 128 ? N_NODES - m0 : 128);

    v8f acc[8] = {};

    for (int kp = 0; kp < IN_F / KP; ++kp) {
        __syncthreads();                    // previous panel fully consumed
        if (wave == 0) {
            // feat panel: tile 32(x) x 128(y), row stride 256; pad 32 DWORDs -> +4 (FS=36)
            tdm_load_2d(lds_off(lds_feat),
                        feat + (long long)m0 * IN_F + kp * KP,
                        /*tensor_d0=*/KP, /*tensor_d1=*/rows_left,
                        /*tile_d0=*/KP, /*tile_d1=*/128,
                        /*d0_stride=*/IN_F, /*pad_code=*/4, /*pad_amt=*/3);
            // weight panel: tile 128(x) x 32(y), row stride 128; pad 128 DWORDs -> +4 (WSTR=132)
            tdm_load_2d(lds_off(lds_w),
                        weight + (long long)(kp * KP) * OUT_F,
                        /*tensor_d0=*/OUT_F, /*tensor_d1=*/KP,
                        /*tile_d0=*/OUT_F, /*tile_d1=*/KP,
                        /*d0_stride=*/OUT_F, /*pad_code=*/6, /*pad_amt=*/3);
            __builtin_amdgcn_s_wait_tensorcnt(0);
        }
        __syncthreads();                    // panel visible to all waves

        #pragma unroll
        for (int kk = 0; kk < KP; kk += 4) {
            v2f a = *(const v2f*)&lds_feat[arow * FS + kk + koff];   // ds_load_b64
            a = a * nrm;                                              // fold norm_src
            #pragma unroll
            for (int t = 0; t < 8; ++t) {
                v2f b;
                b.x = lds_w[(kk + koff)     * WSTR + (t << 4) + l15];
                b.y = lds_w[(kk + koff + 1) * WSTR + (t << 4) + l15];
                acc[t] = __builtin_amdgcn_wmma_f32_16x16x4_f32(
                    /*neg_a=*/false, a, /*neg_b=*/false, b,
                    /*c_mod=*/(short)0, acc[t], /*reuse_a=*/false, /*reuse_b=*/false);
            }
        }
    }

    // ---- store D: VGPR j -> M = j (lanes 0-15) / j+8 (lanes 16-31); N = t*16 + l15
    const int mbase = m0 + (wave << 4) + ((lane >> 4) << 3);
    #pragma unroll
    for (int j = 0; j < 8; ++j) {
        int m = mbase + j;
        if (m < N_NODES) {
            #pragma unroll
            for (int t = 0; t < 8; ++t)
                h[(long long)m * OUT_F + (t << 4) + l15] = acc[t][j];
        }
    }
}

// ---------------------------------------------------------------- edge scatter
// out[dst[e], :] += h[src[e], :]   one wave per edge, float4 per lane.
// h and out are both L2-resident (51.2 MB each << 192 MB L2).
__global__ void scatter_kernel(const int* __restrict__ src, const int* __restrict__ dst,
                               const float* __restrict__ h, float* __restrict__ out) {
    long long tid = (long long)blockIdx.x * blockDim.x + threadIdx.x;
    int e = (int)(tid >> 5);
    if (e >= N_EDGES) return;
    int c = ((int)tid & 31) << 2;                 // 0..124
    int s = src[e];
    int d = dst[e];
    float4 v = *(const float4*)(h + (long long)s * OUT_F + c);
    float* o = out + (long long)d * OUT_F + c;
    atomic_add_f32(o + 0, v.x);
    atomic_add_f32(o + 1, v.y);
    atomic_add_f32(o + 2, v.z);
    atomic_add_f32(o + 3, v.w);
}

// ---------------------------------------------------------------- finalize
__global__ void finalize_kernel(float* __restrict__ out, const float* __restrict__ deg_in,
                                const float* __restrict__ bias) {
    long long tid = (long long)blockIdx.x * blockDim.x + threadIdx.x;
    if (tid >= (long long)N_NODES * OUT_F) return;
    int n = (int)(tid >> 7);
    int f = (int)(tid & 127);
    float d = deg_in[n];
    float nr = rsqrtf(d < 1.0f ? 1.0f : d);
    out[tid] = out[tid] * nr + bias[f];
}

// ---------------------------------------------------------------- launch
extern "C" void kernel_launch(void* const* d_in, const int* in_sizes, int n_in,
                              void* d_out, int out_size, void* d_ws, size_t ws_size,
                              hipStream_t stream) {
    const float* feat   = (const float*)d_in[0];
    const float* weight = (const float*)d_in[1];
    const float* bias   = (const float*)d_in[2];
    const int*   src    = (const int*)d_in[3];
    const int*   dst    = (const int*)d_in[4];
    float*       out    = (float*)d_out;

    float* ws      = (float*)d_ws;
    float* deg_out = ws;                         // N_NODES floats
    float* deg_in  = ws + N_NODES;               // N_NODES floats
    float* h       = ws + 200704;                // N_NODES*OUT_F floats (aligned)

    // zero degree arrays and output accumulator
    zero_kernel<<<1024, 256, 0, stream>>>(ws, 2LL * N_NODES);
    zero_kernel<<<4096, 256, 0, stream>>>(out, (long long)N_NODES * OUT_F);

    // degree histograms
    degree_kernel<<<(N_EDGES + 255) / 256, 256, 0, stream>>>(src, dst, deg_out, deg_in);

    // h = (feat * norm_src) @ W  via f32 WMMA, panels staged by the TDM
    gemm_kernel<<<(N_NODES + 127) / 128, 256, 0, stream>>>(feat, weight, deg_out, h);

    // sparse aggregation
    scatter_kernel<<<(int)(((long long)N_EDGES * 32) / 256), 256, 0, stream>>>(src, dst, h, out);

    // out = agg * norm_dst + bias
    finalize_kernel<<<(int)(((long long)N_NODES * OUT_F + 255) / 256), 256, 0, stream>>>(out, deg_in, bias);
}